// Decoder3_7928509629190
// MI455X (gfx1250) — compile-verified
//
#include <hip/hip_runtime.h>
#include <hip/hip_bf16.h>
#include <float.h>

// Problem dims (from reference)
#define B_   128
#define T_   8
#define E_   52
#define S_   2048
#define R_   512
#define L_   4
#define K2_  (2 * E_)        // 104, divisible by 4
#define SCHUNKS_ 16
#define SCH_ (S_ / SCHUNKS_) // 128

typedef __attribute__((ext_vector_type(2))) float v2f;
typedef __attribute__((ext_vector_type(8))) float v8f;

__device__ __forceinline__ v2f ld2(const float* p) {
    // all call sites are 8-byte aligned (even-float offsets, even row strides)
    return *reinterpret_cast<const v2f*>(p);
}

__device__ __forceinline__ v8f wmma_f32(v2f a, v2f b, v8f c) {
    // D = A(16x4 f32) * B(4x16 f32) + C(16x16 f32), full f32 precision
    return __builtin_amdgcn_wmma_f32_16x16x4_f32(
        /*neg_a=*/false, a, /*neg_b=*/false, b,
        /*c_mod=*/(short)0, c, /*reuse_a=*/false, /*reuse_b=*/false);
}

__device__ __forceinline__ void atomicMaxFloat(float* addr, float v) {
    // Sign-aware trick: positive floats order like int, negative like reversed uint.
    if (v >= 0.0f) atomicMax(reinterpret_cast<int*>(addr), __float_as_int(v));
    else           atomicMin(reinterpret_cast<unsigned int*>(addr), __float_as_uint(v));
}

__device__ __forceinline__ float sigmoidf_(float x) {
    return 1.0f / (1.0f + __expf(-x));
}

// ---------------------------------------------------------------------------
// Kernel 0: init code_pool to -FLT_MAX
// ---------------------------------------------------------------------------
__global__ void init_neg_kernel(float* __restrict__ p, int n) {
    int i = blockIdx.x * blockDim.x + threadIdx.x;
    if (i < n) p[i] = -FLT_MAX;
}

// ---------------------------------------------------------------------------
// Kernel 1: ragged max-pool over S (bandwidth-dominant stage, ~512 MB read)
// grid (B, SCHUNKS), block 128; each thread owns 4 consecutive r (float4)
// ---------------------------------------------------------------------------
__global__ __launch_bounds__(128) void ragged_maxpool_kernel(
    const float* __restrict__ enc, const int* __restrict__ enc_len,
    float* __restrict__ code_pool)
{
    const int b = blockIdx.x;
    const int chunk = blockIdx.y;
    const int len = enc_len[b];
    const int s0 = chunk * SCH_;
    const int s1 = min(len, s0 + SCH_);
    if (s0 >= s1) return;

    const float4* base = reinterpret_cast<const float4*>(enc + (size_t)b * S_ * R_);
    float4 acc = make_float4(-FLT_MAX, -FLT_MAX, -FLT_MAX, -FLT_MAX);
    const int q = threadIdx.x;            // float4 index within a row (R/4 = 128)
    for (int s = s0; s < s1; ++s) {
        float4 v = base[(size_t)s * (R_ / 4) + q];
        acc.x = fmaxf(acc.x, v.x);
        acc.y = fmaxf(acc.y, v.y);
        acc.z = fmaxf(acc.z, v.z);
        acc.w = fmaxf(acc.w, v.w);
    }
    float* dst = code_pool + b * R_ + q * 4;
    atomicMaxFloat(dst + 0, acc.x);
    atomicMaxFloat(dst + 1, acc.y);
    atomicMaxFloat(dst + 2, acc.z);
    atomicMaxFloat(dst + 3, acc.w);
}

// ---------------------------------------------------------------------------
// Kernel 2: a[b,l] = code_pool[b,:] . W_fc_pool[l,:]   (one wave per b)
// ---------------------------------------------------------------------------
__global__ __launch_bounds__(32) void pool_fc_kernel(
    const float* __restrict__ code_pool, const float* __restrict__ W_fc,
    float* __restrict__ a)
{
    const int b = blockIdx.x;
    const int lane = threadIdx.x;
    float s0 = 0.f, s1 = 0.f, s2 = 0.f, s3 = 0.f;
    for (int r = lane; r < R_; r += 32) {
        const float c = code_pool[b * R_ + r];
        s0 += c * W_fc[0 * R_ + r];
        s1 += c * W_fc[1 * R_ + r];
        s2 += c * W_fc[2 * R_ + r];
        s3 += c * W_fc[3 * R_ + r];
    }
    for (int off = 16; off > 0; off >>= 1) {
        s0 += __shfl_xor(s0, off, 32);
        s1 += __shfl_xor(s1, off, 32);
        s2 += __shfl_xor(s2, off, 32);
        s3 += __shfl_xor(s3, off, 32);
    }
    if (lane == 0) {
        a[b * L_ + 0] = s0; a[b * L_ + 1] = s1;
        a[b * L_ + 2] = s2; a[b * L_ + 3] = s3;
    }
}

// ---------------------------------------------------------------------------
// Kernel 3 (per step t): bmi head + output writes + x assembly
// one wave per batch row; block = 128 threads = 4 waves; grid = B/4
// out layout: dec_output[0,1024) dec_prob[1024,5120) disc[5120,9216) kl[9216,10240)
// ---------------------------------------------------------------------------
__global__ __launch_bounds__(128) void ctrl_step_kernel(
    const float* __restrict__ hin, const float* __restrict__ a,
    const float* __restrict__ W_reg, const float* __restrict__ b_reg,
    const float* __restrict__ featEmbed, const float* __restrict__ labelsEmbed,
    const int* __restrict__ mask, const int* __restrict__ /*unused*/,
    const float* __restrict__ emb_table,
    float* __restrict__ xbuf, float* __restrict__ out, int t)
{
    const int lane = threadIdx.x & 31;
    const int wave = threadIdx.x >> 5;
    const int b = blockIdx.x * 4 + wave;

    float s0 = 0.f, s1 = 0.f, s2 = 0.f, s3 = 0.f;
    for (int r = lane; r < R_; r += 32) {
        const float hv = hin[b * R_ + r];
        s0 += hv * W_reg[0 * R_ + r];
        s1 += hv * W_reg[1 * R_ + r];
        s2 += hv * W_reg[2 * R_ + r];
        s3 += hv * W_reg[3 * R_ + r];
    }
    for (int off = 16; off > 0; off >>= 1) {
        s0 += __shfl_xor(s0, off, 32);
        s1 += __shfl_xor(s1, off, 32);
        s2 += __shfl_xor(s2, off, 32);
        s3 += __shfl_xor(s3, off, 32);
    }
    // every lane now holds full sums
    float v[4];
    v[0] = s0 + b_reg[0] + a[b * L_ + 0];
    v[1] = s1 + b_reg[1] + a[b * L_ + 1];
    v[2] = s2 + b_reg[2] + a[b * L_ + 2];
    v[3] = s3 + b_reg[3] + a[b * L_ + 3];

    int label = 0;
    float best = v[0];
    #pragma unroll
    for (int l = 1; l < L_; ++l) {
        if (v[l] > best) { best = v[l]; label = l; }   // first-max (jnp.argmax)
    }
    const int REMAP[4] = {175, 176, 44, 173};
    const int ld = REMAP[label];

    if (lane == 0) {
        const int tb = t * B_ + b;
        out[tb] = (float)label;                    // dec_output
        #pragma unroll
        for (int l = 0; l < L_; ++l) {
            out[1024 + tb * L_ + l] = sigmoidf_(v[l]);  // dec_prob
            out[5120 + tb * L_ + l] = v[l];             // disc_input (pre-sigmoid)
        }
        out[9216 + tb] = (float)ld;                // kl_input
    }

    // x = concat(featEmbed[b,t], bmi) with bmi = mask ? labelsEmbed : emb_table[ld]
    const int m = mask[b * T_ + t];
    for (int e = lane; e < K2_; e += 32) {
        float val;
        if (e < E_) {
            val = featEmbed[((size_t)b * T_ + t) * E_ + e];
        } else {
            const int j = e - E_;
            val = m ? labelsEmbed[((size_t)b * T_ + t) * E_ + j]
                    : emb_table[(size_t)ld * E_ + j];
        }
        xbuf[b * K2_ + e] = val;
    }
}

// ---------------------------------------------------------------------------
// Kernel 4 (per step t): fused GRU cell via V_WMMA_F32_16X16X4_F32
// Each wave computes one 16(M) x 16(N over [0,512)) tile of h_new, keeping
// 4 accumulators: r-sum (gi_r+gh_r), z-sum, i_n, h_n (separate, since
// n = tanh(i_n + r*h_n)).  A-layout: lane(0-15)=M, VGPR pair = K{0,1}/{2,3}
// split by lane>=16 (ISA 7.12.2, 32-bit A 16x4).  B mirrors with N=lane%16.
// grid (8 m-tiles, 8) x 128 threads (4 waves -> 32 c-tiles).
// ---------------------------------------------------------------------------
__global__ __launch_bounds__(128) void gru_step_kernel(
    const float* __restrict__ hin, const float* __restrict__ xbuf,
    const float* __restrict__ W_ih, const float* __restrict__ W_hh,
    const float* __restrict__ b_ih, const float* __restrict__ b_hh,
    float* __restrict__ hout)
{
    const int lane = threadIdx.x & 31;
    const int wave = threadIdx.x >> 5;
    const int m0 = blockIdx.x << 4;
    const int c0 = ((blockIdx.y << 2) + wave) << 4;
    const int arow = m0 + (lane & 15);        // A-matrix row fed by this lane
    const int koff = (lane >> 4) << 1;        // K sub-offset {0,2}
    const int n    = c0 + (lane & 15);        // B-matrix column fed by this lane

    v8f accR = {}; v8f accZ = {}; v8f accIN = {}; v8f accHN = {};

    // ---- gh = h @ W_hh.T, K = 512 ----
    const float* Ap  = hin  + (size_t)arow * R_ + koff;
    const float* Brp = W_hh + (size_t)(n + 0 * R_) * R_ + koff;
    const float* Bzp = W_hh + (size_t)(n + 1 * R_) * R_ + koff;
    const float* Bnp = W_hh + (size_t)(n + 2 * R_) * R_ + koff;
    #pragma unroll 4
    for (int k = 0; k < R_; k += 4) {
        const v2f av = ld2(Ap + k);
        accR  = wmma_f32(av, ld2(Brp + k), accR);
        accZ  = wmma_f32(av, ld2(Bzp + k), accZ);
        accHN = wmma_f32(av, ld2(Bnp + k), accHN);
    }

    // ---- gi = x @ W_ih.T, K = 104 ----
    const float* Xp  = xbuf + (size_t)arow * K2_ + koff;
    const float* Crp = W_ih + (size_t)(n + 0 * R_) * K2_ + koff;
    const float* Czp = W_ih + (size_t)(n + 1 * R_) * K2_ + koff;
    const float* Cnp = W_ih + (size_t)(n + 2 * R_) * K2_ + koff;
    #pragma unroll 4
    for (int k = 0; k < K2_; k += 4) {
        const v2f av = ld2(Xp + k);
        accR  = wmma_f32(av, ld2(Crp + k), accR);
        accZ  = wmma_f32(av, ld2(Czp + k), accZ);
        accIN = wmma_f32(av, ld2(Cnp + k), accIN);
    }

    // ---- gates + state update ----
    const float bR = b_ih[n + 0 * R_] + b_hh[n + 0 * R_];
    const float bZ = b_ih[n + 1 * R_] + b_hh[n + 1 * R_];
    const float bI = b_ih[n + 2 * R_];
    const float bH = b_hh[n + 2 * R_];
    const int mbase = m0 + ((lane >> 4) << 3);   // C/D layout: lanes>=16 hold M=i+8
    #pragma unroll
    for (int i = 0; i < 8; ++i) {
        const int m = mbase + i;
        const float r  = sigmoidf_(accR[i] + bR);
        const float z  = sigmoidf_(accZ[i] + bZ);
        const float nn = tanhf(accIN[i] + bI + r * (accHN[i] + bH));
        const float hp = hin[(size_t)m * R_ + n];
        hout[(size_t)m * R_ + n] = (1.0f - z) * nn + z * hp;
    }
}

// ---------------------------------------------------------------------------
// Host-side launch sequence (graph-capture safe: only kernels + async d2d copy)
// ---------------------------------------------------------------------------
extern "C" void kernel_launch(void* const* d_in, const int* in_sizes, int n_in,
                              void* d_out, int out_size, void* d_ws, size_t ws_size,
                              hipStream_t stream) {
    (void)in_sizes; (void)n_in; (void)out_size; (void)ws_size;

    const float* featEmbed   = (const float*)d_in[0];
    const float* labelsEmbed = (const float*)d_in[1];
    const float* enc         = (const float*)d_in[2];
    const float* h_n         = (const float*)d_in[3];
    const int*   mask        = (const int*)  d_in[4];
    const int*   enc_len     = (const int*)  d_in[5];
    const float* emb_table   = (const float*)d_in[6];
    const float* W_fc        = (const float*)d_in[7];
    const float* W_reg       = (const float*)d_in[8];
    const float* b_reg       = (const float*)d_in[9];
    const float* W_ih        = (const float*)d_in[10];
    const float* W_hh        = (const float*)d_in[11];
    const float* b_ih        = (const float*)d_in[12];
    const float* b_hh        = (const float*)d_in[13];

    float* out = (float*)d_out;
    float* ws  = (float*)d_ws;

    float* code_pool = ws;                  // 65536 floats
    float* a         = ws + 65536;          //   512
    float* h0        = ws + 66048;          // 65536
    float* h1        = ws + 131584;         // 65536
    float* xbuf      = ws + 197120;         // 13312  (total ~0.8 MB)

    // Stage 1: ragged max-pool (bandwidth-dominant) + tiny FC
    init_neg_kernel<<<(B_ * R_ + 255) / 256, 256, 0, stream>>>(code_pool, B_ * R_);
    ragged_maxpool_kernel<<<dim3(B_, SCHUNKS_), 128, 0, stream>>>(enc, enc_len, code_pool);
    pool_fc_kernel<<<B_, 32, 0, stream>>>(code_pool, W_fc, a);

    // Stage 2: seed h, then 8 sequential GRU steps (ping-pong h buffers)
    hipMemcpyAsync(h0, h_n, (size_t)B_ * R_ * sizeof(float),
                   hipMemcpyDeviceToDevice, stream);
    float* hb[2] = {h0, h1};
    for (int t = 0; t < T_; ++t) {
        ctrl_step_kernel<<<B_ / 4, 128, 0, stream>>>(
            hb[t & 1], a, W_reg, b_reg, featEmbed, labelsEmbed,
            mask, nullptr, emb_table, xbuf, out, t);
        gru_step_kernel<<<dim3(8, 8), 128, 0, stream>>>(
            hb[t & 1], xbuf, W_ih, W_hh, b_ih, b_hh, hb[(t + 1) & 1]);
    }
}